// GATLayer_420906795779
// MI455X (gfx1250) — compile-verified
//
#include <hip/hip_runtime.h>
#include <hip/hip_bf16.h>

#define N_NODES 50000
#define N_EDGES 800000
#define IN_DIM 128
#define OUT_DIM 64
#define NEG_SLOPE 0.01f

typedef __attribute__((ext_vector_type(2))) float v2f;
typedef __attribute__((ext_vector_type(8))) float v8f;

// ---------------------------------------------------------------------------
// Kernel 1: z = h @ W_fc^T  via V_WMMA_F32_16X16X4_F32 (fp32 exact, BW-bound)
// grid.x = 3125 (16-row tiles), block = 128 threads (4 waves; wave w owns the
// 16-col N-tile w). A-tile (16x128) staged in LDS, padded stride 132.
// ---------------------------------------------------------------------------
__global__ __launch_bounds__(128) void gat_gemm_z(
    const float* __restrict__ h, const float* __restrict__ Wfc,
    float* __restrict__ z)
{
    __shared__ float Atile[16 * 132];

    const int tid   = threadIdx.x;
    const int mbase = blockIdx.x * 16;

    // Cooperative coalesced load of the 16x128 fp32 A tile (float4 = 512 elts)
    for (int i = tid; i < 16 * 32; i += 128) {
        const int r  = i >> 5;
        const int c4 = i & 31;
        float4 v = ((const float4*)(h + (size_t)(mbase + r) * IN_DIM))[c4];
        *(float4*)&Atile[r * 132 + c4 * 4] = v;
    }
    __syncthreads();

    const int wave  = tid >> 5;        // N-tile (0..3)
    const int lane  = tid & 31;
    const int nloc  = lane & 15;       // column within 16-wide tile
    const int half  = lane >> 4;       // 0: K pair {0,1}/rows 0-7, 1: {2,3}/rows 8-15
    const int n     = wave * 16 + nloc;
    const int mloc  = lane & 15;       // A fragment row

    v8f c = {0.f, 0.f, 0.f, 0.f, 0.f, 0.f, 0.f, 0.f};

#pragma unroll
    for (int k = 0; k < IN_DIM; k += 4) {
        const int ka = k + half * 2;
        // A fragment: lanes 0-15 rows 0-15 (K=ka,ka+1), lanes 16-31 same rows (K=ka+2..)
        v2f a = *(const v2f*)&Atile[mloc * 132 + ka];
        // B fragment: B[k'][n] = Wfc[n][k']  (B = W_fc^T)
        v2f b;
        b.x = Wfc[n * IN_DIM + ka];
        b.y = Wfc[n * IN_DIM + ka + 1];
        c = __builtin_amdgcn_wmma_f32_16x16x4_f32(
                /*neg_a=*/false, a, /*neg_b=*/false, b,
                /*c_mod=*/(short)0, c, /*reuse_a=*/false, /*reuse_b=*/false);
    }

    // C/D layout: lane holds column n, rows mbase + half*8 + j  (j = VGPR idx)
#pragma unroll
    for (int j = 0; j < 8; ++j) {
        z[(size_t)(mbase + half * 8 + j) * OUT_DIM + n] = c[j];
    }
}

// ---------------------------------------------------------------------------
// Kernel 2: per-node prep: s_src = z . a_src, s_dst = z . a_dst,
//           m = -inf, denom = 0, out row = 0.
// ---------------------------------------------------------------------------
__global__ __launch_bounds__(256) void gat_node_prep(
    const float* __restrict__ z, const float* __restrict__ Wattn,
    float* __restrict__ ssrc, float* __restrict__ sdst,
    float* __restrict__ mmax, float* __restrict__ denom,
    float* __restrict__ out)
{
    const int i = blockIdx.x * blockDim.x + threadIdx.x;
    if (i >= N_NODES) return;

    const float4* zr = (const float4*)(z + (size_t)i * OUT_DIM);
    const float4* as = (const float4*)(Wattn);            // a_src = W_attn[0, :64]
    const float4* ad = (const float4*)(Wattn + OUT_DIM);  // a_dst = W_attn[0, 64:]

    float accs = 0.f, accd = 0.f;
#pragma unroll
    for (int j = 0; j < OUT_DIM / 4; ++j) {
        float4 v = zr[j];
        float4 s = as[j];
        float4 d = ad[j];
        accs += v.x * s.x + v.y * s.y + v.z * s.z + v.w * s.w;
        accd += v.x * d.x + v.y * d.y + v.z * d.z + v.w * d.w;
    }
    ssrc[i]  = accs;
    sdst[i]  = accd;
    mmax[i]  = -INFINITY;
    denom[i] = 0.f;

    float4 zero = {0.f, 0.f, 0.f, 0.f};
    float4* orow = (float4*)(out + (size_t)i * OUT_DIM);
#pragma unroll
    for (int j = 0; j < OUT_DIM / 4; ++j) orow[j] = zero;
}

// ---------------------------------------------------------------------------
// Kernel 3: per-edge score + leaky relu + segment max (native f32 atomic max)
// ---------------------------------------------------------------------------
__global__ __launch_bounds__(256) void gat_edge_score(
    const int* __restrict__ src, const int* __restrict__ dst,
    const float* __restrict__ ssrc, const float* __restrict__ sdst,
    float* __restrict__ mmax, float* __restrict__ ew)
{
    const int i = blockIdx.x * blockDim.x + threadIdx.x;
    if (i >= N_EDGES) return;
    const int s = src[i];
    const int d = dst[i];
    float e = ssrc[s] + sdst[d];
    e = (e >= 0.f) ? e : NEG_SLOPE * e;
    ew[i] = e;
    unsafeAtomicMax(&mmax[d], e);
}

// ---------------------------------------------------------------------------
// Kernel 4: exp(e - m[dst]) and segment-sum denominator
// ---------------------------------------------------------------------------
__global__ __launch_bounds__(256) void gat_edge_exp(
    const int* __restrict__ dst, const float* __restrict__ mmax,
    float* __restrict__ ew, float* __restrict__ denom)
{
    const int i = blockIdx.x * blockDim.x + threadIdx.x;
    if (i >= N_EDGES) return;
    const int d = dst[i];
    const float x = __expf(ew[i] - mmax[d]);
    ew[i] = x;
    unsafeAtomicAdd(&denom[d], x);
}

// ---------------------------------------------------------------------------
// Kernel 5: out[dst] += alpha * z[src]   (wave32 per edge, float2 per lane)
// grid = N_EDGES/8 blocks of 256 threads (8 waves/block).
// ---------------------------------------------------------------------------
__global__ __launch_bounds__(256) void gat_edge_aggregate(
    const int* __restrict__ src, const int* __restrict__ dst,
    const float* __restrict__ z, const float* __restrict__ ew,
    const float* __restrict__ denom, float* __restrict__ out)
{
    const int edge = blockIdx.x * 8 + (threadIdx.x >> 5);
    const int lane = threadIdx.x & 31;
    const int s = src[edge];
    const int d = dst[edge];
    const float alpha = ew[edge] / denom[d];

    const float2 v = ((const float2*)(z + (size_t)s * OUT_DIM))[lane];
    float* orow = out + (size_t)d * OUT_DIM + 2 * lane;
    unsafeAtomicAdd(orow + 0, alpha * v.x);
    unsafeAtomicAdd(orow + 1, alpha * v.y);
}

// ---------------------------------------------------------------------------
extern "C" void kernel_launch(void* const* d_in, const int* in_sizes, int n_in,
                              void* d_out, int out_size, void* d_ws, size_t ws_size,
                              hipStream_t stream) {
    const float* h     = (const float*)d_in[0];
    const int*   src   = (const int*)  d_in[1];
    const int*   dst   = (const int*)  d_in[2];
    const float* Wfc   = (const float*)d_in[3];
    const float* Wattn = (const float*)d_in[4];
    float* out = (float*)d_out;

    // Workspace layout (bytes):
    char* ws = (char*)d_ws;
    float* z     = (float*)(ws);                      // 50000*64*4 = 12,800,000
    float* ssrc  = (float*)(ws + 12800000);           // 200,000
    float* sdst  = (float*)(ws + 13000000);           // 200,000
    float* mmax  = (float*)(ws + 13200000);           // 200,000
    float* denom = (float*)(ws + 13400000);           // 200,000
    float* ew    = (float*)(ws + 13600000);           // 800000*4 = 3,200,000

    // 1) z = h @ W_fc^T  (WMMA f32)
    gat_gemm_z<<<N_NODES / 16, 128, 0, stream>>>(h, Wfc, z);

    // 2) per-node scores + init m/denom/out
    gat_node_prep<<<(N_NODES + 255) / 256, 256, 0, stream>>>(
        z, Wattn, ssrc, sdst, mmax, denom, out);

    // 3) edge scores + segment max
    gat_edge_score<<<N_EDGES / 256, 256, 0, stream>>>(
        src, dst, ssrc, sdst, mmax, ew);

    // 4) exp + denominator
    gat_edge_exp<<<N_EDGES / 256, 256, 0, stream>>>(dst, mmax, ew, denom);

    // 5) weighted aggregation (wave per edge)
    gat_edge_aggregate<<<N_EDGES / 8, 256, 0, stream>>>(
        src, dst, z, ew, denom, out);
}